// ProbAttention_83210696392949
// MI455X (gfx1250) — compile-verified
//
#include <hip/hip_runtime.h>
#include <hip/hip_bf16.h>

// ---------------- problem constants (from reference) ----------------
#define BATCH 4
#define SEQL  2048
#define DMODEL 512
#define HEADS 8
#define EDIM  64            // DMODEL / HEADS
#define UQ    40            // FACTOR * ceil(log(L)) = 5*8
#define BH    (BATCH*HEADS) // 32
#define MROWS (BATCH*SEQL)  // 8192 rows for projection GEMMs
#define SCALE 0.125f        // 1/sqrt(EDIM)
#define KC    64            // K-chunk staged in LDS
#define WPAD  72            // padded row so half-wave K-groups use disjoint banks
#define NSEG  8
#define SEGLEN (SEQL/NSEG)  // 256

typedef float v2f __attribute__((ext_vector_type(2)));
typedef float v8f __attribute__((ext_vector_type(8)));
typedef int   v4i_g __attribute__((vector_size(16)));   // matches builtin param type

#define AS1 __attribute__((address_space(1)))
#define AS3 __attribute__((address_space(3)))

#if __has_builtin(__builtin_amdgcn_global_load_async_to_lds_b128) && \
    __has_builtin(__builtin_amdgcn_s_wait_asynccnt)
#define USE_ASYNC_LDS 1
#else
#define USE_ASYNC_LDS 0
#endif

// =====================================================================
// 1) Projection GEMM:  C[M=8192, N=512] = A[8192x512] * W[512x512]
//    fp32 WMMA 16x16x4.  Wave: 32 rows x 64 cols (8 accumulators).
//    W strip staged through double-buffered LDS chunks (async if available):
//    per k-step = 2 global b64 (A) + 8 ds b32 (W) for 8 WMMAs.
//    Grid: (M/256, N/64), block 256 = 8 waves.
// =====================================================================
__global__ __launch_bounds__(256) void proj_gemm(const float* __restrict__ A,
                                                 const float* __restrict__ W,
                                                 float* __restrict__ C) {
  __shared__ float wt[2][KC][WPAD];          // 2 x 18 KB
  const int tid  = threadIdx.x;
  const int wave = tid >> 5;
  const int lane = tid & 31;
  const int l15  = lane & 15;
  const int khalf = (lane >> 4) << 1;        // 0 / 2
  const int row0 = (blockIdx.x * 8 + wave) * 32;
  const int col0 = blockIdx.y * 64;

  // stage one 64x64 W chunk into wt[buf]: 4 x b128 per thread
  auto stage = [&](int buf, int kcbase) {
#pragma unroll
    for (int j = 0; j < 4; ++j) {
      const int f = tid + 256 * j;           // 0..1023 float4 slots
      const int r = f >> 4;                  // k-row within chunk
      const int c = (f & 15) * 4;            // col within strip
      const float* src = W + (size_t)(kcbase + r) * DMODEL + col0 + c;
      float* dst = &wt[buf][r][c];
#if USE_ASYNC_LDS
      __builtin_amdgcn_global_load_async_to_lds_b128(
          (AS1 v4i_g*)(uintptr_t)src,
          (AS3 v4i_g*)(uint32_t)(uintptr_t)dst, 0, 0);
#else
      *(float4*)dst = *(const float4*)src;
#endif
    }
  };

  const float* a0row = A + (size_t)(row0 + l15) * DMODEL;
  const float* a1row = A + (size_t)(row0 + 16 + l15) * DMODEL;

  v8f acc00 = {}, acc01 = {}, acc02 = {}, acc03 = {};
  v8f acc10 = {}, acc11 = {}, acc12 = {}, acc13 = {};

  int buf = 0;
  stage(0, 0);
#if USE_ASYNC_LDS
  __builtin_amdgcn_s_wait_asynccnt(0);
#endif
  __syncthreads();

  for (int ch = 0; ch < DMODEL / KC; ++ch) {
    if (ch + 1 < DMODEL / KC) stage(buf ^ 1, (ch + 1) * KC);
    const int kcbase = ch * KC;
#pragma unroll 4
    for (int kk = 0; kk < KC; kk += 4) {
      const int kb = kk + khalf;
      v2f a0 = *(const v2f*)(a0row + kcbase + kb);
      v2f a1 = *(const v2f*)(a1row + kcbase + kb);
      const float* wr0 = &wt[buf][kb][l15];
      const float* wr1 = &wt[buf][kb + 1][l15];
      v2f b0; b0.x = wr0[0];  b0.y = wr1[0];
      v2f b1; b1.x = wr0[16]; b1.y = wr1[16];
      v2f b2; b2.x = wr0[32]; b2.y = wr1[32];
      v2f b3; b3.x = wr0[48]; b3.y = wr1[48];
      acc00 = __builtin_amdgcn_wmma_f32_16x16x4_f32(false, a0, false, b0, (short)0, acc00, false, false);
      acc01 = __builtin_amdgcn_wmma_f32_16x16x4_f32(false, a0, false, b1, (short)0, acc01, false, false);
      acc02 = __builtin_amdgcn_wmma_f32_16x16x4_f32(false, a0, false, b2, (short)0, acc02, false, false);
      acc03 = __builtin_amdgcn_wmma_f32_16x16x4_f32(false, a0, false, b3, (short)0, acc03, false, false);
      acc10 = __builtin_amdgcn_wmma_f32_16x16x4_f32(false, a1, false, b0, (short)0, acc10, false, false);
      acc11 = __builtin_amdgcn_wmma_f32_16x16x4_f32(false, a1, false, b1, (short)0, acc11, false, false);
      acc12 = __builtin_amdgcn_wmma_f32_16x16x4_f32(false, a1, false, b2, (short)0, acc12, false, false);
      acc13 = __builtin_amdgcn_wmma_f32_16x16x4_f32(false, a1, false, b3, (short)0, acc13, false, false);
    }
#if USE_ASYNC_LDS
    __builtin_amdgcn_s_wait_asynccnt(0);
#endif
    __syncthreads();
    buf ^= 1;
  }

  const int rb = (lane >> 4) << 3;
#pragma unroll
  for (int r = 0; r < 8; ++r) {
    float* c0 = C + (size_t)(row0 + rb + r) * DMODEL + col0 + l15;
    c0[0]  = acc00[r]; c0[16] = acc01[r]; c0[32] = acc02[r]; c0[48] = acc03[r];
    float* c1 = C + (size_t)(row0 + 16 + rb + r) * DMODEL + col0 + l15;
    c1[0]  = acc10[r]; c1[16] = acc11[r]; c1[32] = acc12[r]; c1[48] = acc13[r];
  }
}

// =====================================================================
// 2) Sampled scores -> m[b,h,l] = max_s(q.k_s) - sum_s(q.k_s)/L
// =====================================================================
__global__ __launch_bounds__(256) void sample_m_kernel(const float* __restrict__ qp,
                                                       const float* __restrict__ kp,
                                                       const int* __restrict__ sidx,
                                                       float* __restrict__ m) {
  const int gid = blockIdx.x * blockDim.x + threadIdx.x;   // over BH*L
  const int l  = gid & (SEQL - 1);
  const int bh = gid >> 11;
  const float4* q4 = (const float4*)(qp + ((size_t)bh * SEQL + l) * EDIM);
  float4 qv[EDIM / 4];
#pragma unroll
  for (int i = 0; i < EDIM / 4; ++i) qv[i] = q4[i];

  float mx = -3.402823466e+38f, sm = 0.f;
  for (int s = 0; s < UQ; ++s) {
    const int kl = sidx[l * UQ + s];
    const float4* k4 = (const float4*)(kp + ((size_t)bh * SEQL + kl) * EDIM);
    float acc = 0.f;
#pragma unroll
    for (int i = 0; i < EDIM / 4; ++i) {
      float4 kv = k4[i];
      acc += qv[i].x * kv.x + qv[i].y * kv.y + qv[i].z * kv.z + qv[i].w * kv.w;
    }
    mx = fmaxf(mx, acc);
    sm += acc;
  }
  m[gid] = mx - sm * (1.0f / (float)SEQL);
}

// =====================================================================
// 3) Top-40 per (b,h): iterative extraction, ties -> lower index (jax order)
// =====================================================================
__global__ __launch_bounds__(256) void topk40_kernel(const float* __restrict__ m,
                                                     int* __restrict__ mtop) {
  __shared__ float vals[SEQL];
  __shared__ float cv[256];
  __shared__ int   ci[256];
  const int bh = blockIdx.x;
  const int tid = threadIdx.x;
  for (int i = tid; i < SEQL; i += 256) vals[i] = m[(size_t)bh * SEQL + i];
  __syncthreads();
  for (int u = 0; u < UQ; ++u) {
    float bv = -3.402823466e+38f; int bi = 0x7fffffff;
    for (int i = tid; i < SEQL; i += 256) {
      float v = vals[i];
      if (v > bv || (v == bv && i < bi)) { bv = v; bi = i; }
    }
    cv[tid] = bv; ci[tid] = bi;
    __syncthreads();
    for (int s = 128; s > 0; s >>= 1) {
      if (tid < s) {
        float v2 = cv[tid + s]; int i2 = ci[tid + s];
        if (v2 > cv[tid] || (v2 == cv[tid] && i2 < ci[tid])) { cv[tid] = v2; ci[tid] = i2; }
      }
      __syncthreads();
    }
    if (tid == 0) { mtop[bh * UQ + u] = ci[0]; vals[ci[0]] = -3.402823466e+38f; }
    __syncthreads();
  }
}

// =====================================================================
// 4) scores[u, l] = SCALE * (q_reduce @ k_^T), via WMMA per (b,h).
// =====================================================================
__global__ __launch_bounds__(256) void scores_wmma(const float* __restrict__ qp,
                                                   const float* __restrict__ kp,
                                                   const int* __restrict__ mtop,
                                                   float* __restrict__ scores) {
  __shared__ float qt[48 * EDIM];           // 12 KB
  const int bh  = blockIdx.x;
  const int tid = threadIdx.x;
  for (int i = tid; i < 48 * EDIM; i += 256) {
    const int u = i >> 6, e = i & 63;
    qt[i] = (u < UQ) ? qp[((size_t)bh * SEQL + mtop[bh * UQ + u]) * EDIM + e] : 0.f;
  }
  __syncthreads();

  const int wave = tid >> 5;
  const int lane = tid & 31;
  const int l15  = lane & 15;
  const int khalf = (lane >> 4) << 1;
  const float* kbh = kp + (size_t)bh * SEQL * EDIM;

  for (int t = wave; t < 3 * (SEQL / 16); t += 8) {
    const int tm = t % 3;
    const int tn = t / 3;
    const int col = tn * 16 + l15;          // sequence position (N dim)
    const float* afrag = &qt[(tm * 16 + l15) * EDIM];
    const float* bcol  = kbh + (size_t)col * EDIM;   // B[k,col] = k_[col*E + k]
    v8f acc = {};
#pragma unroll
    for (int kk = 0; kk < EDIM; kk += 4) {
      const int kb = kk + khalf;
      v2f a = *(const v2f*)(afrag + kb);
      v2f b = *(const v2f*)(bcol + kb);
      acc = __builtin_amdgcn_wmma_f32_16x16x4_f32(false, a, false, b, (short)0, acc, false, false);
    }
    const int rbase = tm * 16 + ((lane >> 4) << 3);
    for (int r = 0; r < 8; ++r) {
      const int u = rbase + r;
      if (u < UQ)
        scores[((size_t)bh * UQ + u) * SEQL + tn * 16 + l15] = acc[r] * SCALE;
    }
  }
}

// =====================================================================
// 5) Masked softmax over l <= top; zeros elsewhere.  Block per row.
// =====================================================================
__global__ __launch_bounds__(256) void softmax_rows(float* __restrict__ sc,
                                                    const int* __restrict__ mtop) {
  __shared__ float red[256];
  const int row = blockIdx.x;               // bh*UQ + u
  const int n = mtop[row] + 1;              // valid length (l <= top)
  float* s = sc + (size_t)row * SEQL;
  const int tid = threadIdx.x;

  float mx = -3.402823466e+38f;
  for (int i = tid; i < n; i += 256) mx = fmaxf(mx, s[i]);
  red[tid] = mx; __syncthreads();
  for (int st = 128; st > 0; st >>= 1) {
    if (tid < st) red[tid] = fmaxf(red[tid], red[tid + st]);
    __syncthreads();
  }
  mx = red[0]; __syncthreads();

  float sum = 0.f;
  for (int i = tid; i < n; i += 256) { float e = __expf(s[i] - mx); s[i] = e; sum += e; }
  red[tid] = sum; __syncthreads();
  for (int st = 128; st > 0; st >>= 1) {
    if (tid < st) red[tid] += red[tid + st];
    __syncthreads();
  }
  const float inv = 1.0f / red[0];
  for (int i = tid; i < n; i += 256) s[i] *= inv;
  for (int i = n + tid; i < SEQL; i += 256) s[i] = 0.f;
}

// =====================================================================
// 6) ctx_upd[u, e] = attn(40x2048) @ v_(2048x64), WMMA per (b,h).
// =====================================================================
__global__ __launch_bounds__(256) void ctx_wmma(const float* __restrict__ attn,
                                                const float* __restrict__ vp,
                                                float* __restrict__ ctxu) {
  const int bh   = blockIdx.x;
  const int tid  = threadIdx.x;
  const int wave = tid >> 5;
  const int lane = tid & 31;
  const int l15  = lane & 15;
  const int khalf = (lane >> 4) << 1;
  const float* att = attn + (size_t)bh * UQ * SEQL;
  const float* vbh = vp + (size_t)bh * SEQL * EDIM;

  for (int t = wave; t < 3 * (EDIM / 16); t += 8) {   // 12 tiles
    const int tm = t % 3;
    const int tn = t / 3;
    const int arow = tm * 16 + l15;
    const bool av = arow < UQ;
    const float* ap = att + (size_t)arow * SEQL;
    const int col = tn * 16 + l15;
    v8f accA = {}, accB = {};
    for (int kk = 0; kk < SEQL; kk += 8) {
      const int kb0 = kk + khalf;
      const int kb1 = kb0 + 4;
      v2f a0 = {}, a1 = {};
      if (av) { a0 = *(const v2f*)(ap + kb0); a1 = *(const v2f*)(ap + kb1); }
      v2f b0; b0.x = vbh[(size_t)kb0 * EDIM + col];  b0.y = vbh[(size_t)(kb0 + 1) * EDIM + col];
      v2f b1; b1.x = vbh[(size_t)kb1 * EDIM + col];  b1.y = vbh[(size_t)(kb1 + 1) * EDIM + col];
      accA = __builtin_amdgcn_wmma_f32_16x16x4_f32(false, a0, false, b0, (short)0, accA, false, false);
      accB = __builtin_amdgcn_wmma_f32_16x16x4_f32(false, a1, false, b1, (short)0, accB, false, false);
    }
    v8f acc = accA + accB;
    const int rbase = tm * 16 + ((lane >> 4) << 3);
    for (int r = 0; r < 8; ++r) {
      const int u = rbase + r;
      if (u < UQ)
        ctxu[((size_t)bh * UQ + u) * EDIM + tn * 16 + l15] = acc[r];
    }
  }
}

// =====================================================================
// 7) cumsum(v_, axis=L) -> d_out, two-phase segmented scan.
// =====================================================================
__global__ __launch_bounds__(256) void vseg_sum(const float* __restrict__ vp,
                                                float* __restrict__ segsum) {
  const int gid = blockIdx.x * blockDim.x + threadIdx.x;   // BH*NSEG*EDIM
  const int e   = gid & (EDIM - 1);
  const int seg = (gid >> 6) & (NSEG - 1);
  const int bh  = gid >> 9;
  const float* src = vp + (size_t)bh * SEQL * EDIM + (size_t)seg * SEGLEN * EDIM + e;
  float s = 0.f;
  for (int l = 0; l < SEGLEN; ++l) s += src[(size_t)l * EDIM];
  segsum[gid] = s;
}

__global__ __launch_bounds__(256) void vseg_scan(const float* __restrict__ vp,
                                                 const float* __restrict__ segsum,
                                                 float* __restrict__ out) {
  const int gid = blockIdx.x * blockDim.x + threadIdx.x;   // BH*NSEG*EDIM
  const int e   = gid & (EDIM - 1);
  const int seg = (gid >> 6) & (NSEG - 1);
  const int bh  = gid >> 9;
  float run = 0.f;
  for (int s = 0; s < NSEG; ++s)
    if (s < seg) run += segsum[((size_t)bh * NSEG + s) * EDIM + e];
  const size_t base = (size_t)bh * SEQL * EDIM + (size_t)seg * SEGLEN * EDIM + e;
  const float* src = vp + base;
  float* dst = out + base;
  for (int l = 0; l < SEGLEN; ++l) {
    run += src[(size_t)l * EDIM];
    dst[(size_t)l * EDIM] = run;
  }
}

// 8) scatter the 40 updated rows into the cumsum context
__global__ __launch_bounds__(256) void scatter_kernel(const float* __restrict__ ctxu,
                                                      const int* __restrict__ mtop,
                                                      float* __restrict__ out) {
  const int gid = blockIdx.x * blockDim.x + threadIdx.x;   // BH*UQ*EDIM
  const int e  = gid & (EDIM - 1);
  const int uu = gid >> 6;           // bh*UQ + u
  const int bh = uu / UQ;
  const int l  = mtop[uu];
  out[((size_t)bh * SEQL + l) * EDIM + e] = ctxu[(size_t)uu * EDIM + e];
}

// =====================================================================
extern "C" void kernel_launch(void* const* d_in, const int* in_sizes, int n_in,
                              void* d_out, int out_size, void* d_ws, size_t ws_size,
                              hipStream_t stream) {
  (void)in_sizes; (void)n_in; (void)out_size; (void)ws_size;
  const float* q  = (const float*)d_in[0];
  const float* k  = (const float*)d_in[1];
  const float* v  = (const float*)d_in[2];
  const float* Wq = (const float*)d_in[3];
  const float* Wk = (const float*)d_in[4];
  const float* Wv = (const float*)d_in[5];
  const int* sidx = (const int*)d_in[6];
  float* out = (float*)d_out;

  char* ws = (char*)d_ws;
  size_t off = 0;
  auto take = [&](size_t bytes) -> void* {
    void* p = ws + off;
    off += (bytes + 255) & ~(size_t)255;
    return p;
  };
  float* qp     = (float*)take((size_t)MROWS * DMODEL * 4);    // 16 MB
  float* kp     = (float*)take((size_t)MROWS * DMODEL * 4);    // 16 MB
  float* vp     = (float*)take((size_t)MROWS * DMODEL * 4);    // 16 MB
  float* mbuf   = (float*)take((size_t)BH * SEQL * 4);         // 256 KB
  int*   mtop   = (int*)  take((size_t)BH * UQ * 4);           // 5 KB
  float* scores = (float*)take((size_t)BH * UQ * SEQL * 4);    // 10 MB
  float* ctxu   = (float*)take((size_t)BH * UQ * EDIM * 4);    // 320 KB
  float* segs   = (float*)take((size_t)BH * NSEG * EDIM * 4);  // 64 KB

  dim3 gproj(MROWS / 256, DMODEL / 64);
  proj_gemm<<<gproj, 256, 0, stream>>>(q, Wq, qp);
  proj_gemm<<<gproj, 256, 0, stream>>>(k, Wk, kp);
  proj_gemm<<<gproj, 256, 0, stream>>>(v, Wv, vp);

  sample_m_kernel<<<(BH * SEQL) / 256, 256, 0, stream>>>(qp, kp, sidx, mbuf);
  topk40_kernel<<<BH, 256, 0, stream>>>(mbuf, mtop);
  scores_wmma<<<BH, 256, 0, stream>>>(qp, kp, mtop, scores);
  softmax_rows<<<BH * UQ, 256, 0, stream>>>(scores, mtop);
  ctx_wmma<<<BH, 256, 0, stream>>>(scores, vp, ctxu);
  vseg_sum<<<(BH * NSEG * EDIM) / 256, 256, 0, stream>>>(vp, segs);
  vseg_scan<<<(BH * NSEG * EDIM) / 256, 256, 0, stream>>>(vp, segs, out);
  scatter_kernel<<<(BH * UQ * EDIM) / 256, 256, 0, stream>>>(ctxu, mtop, out);
}